// AttentionHead_38646115729769
// MI455X (gfx1250) — compile-verified
//
#include <hip/hip_runtime.h>

// ---------------------------------------------------------------------------
// AttentionHead (B=4, T=4096, E=1024, H=64), fp32, causal, softmax/sqrt(H) quirk.
// Strategy: fp32 WMMA (v_wmma_f32_16x16x4_f32) for all matmuls, flash-attention
// streaming so the T x T score matrix is never materialized. Q/K/V (12 MB) live
// in d_ws. Compute-bound (~400 flop/byte), K/V stay L2-resident per batch.
// ---------------------------------------------------------------------------

typedef __attribute__((ext_vector_type(2))) float v2f;
typedef __attribute__((ext_vector_type(8))) float v8f;

constexpr int EMB = 1024;
constexpr int HS  = 64;
constexpr int TT  = 4096;
constexpr int BB  = 4;
constexpr int BT  = BB * TT;          // 16384 rows
constexpr size_t QKV1 = (size_t)BT * HS;   // elements per Q/K/V plane

// One fp32 WMMA K-step for the projection: A row chunk (float2) x W column chunk.
__device__ __forceinline__ v8f proj_step(const float* __restrict__ arow,
                                         const float* __restrict__ wcol,
                                         int kk, v8f acc) {
    v2f a = *(const v2f*)(arow + kk);        // A[m][kk], A[m][kk+1]
    v2f b;
    b.x = wcol[(size_t)kk * HS];             // W[kk][n]
    b.y = wcol[(size_t)(kk + 1) * HS];       // W[kk+1][n]
    return __builtin_amdgcn_wmma_f32_16x16x4_f32(false, a, false, b,
                                                 (short)0, acc, false, false);
}

// grid: (BT/16, 12) ; block: 32 (one wave). blockIdx.y: [mat:2bits<<2 | coltile]
__global__ __launch_bounds__(32)
void qkv_proj(const float* __restrict__ x,
              const float* __restrict__ Wq,
              const float* __restrict__ Wk,
              const float* __restrict__ Wv,
              float* __restrict__ qkv) {
    const int lane = threadIdx.x & 31;
    const int nl = lane & 15;        // N / M sub-index
    const int h  = lane >> 4;        // half select -> K offset 2h
    const int rt  = blockIdx.x;      // 16-row tile over B*T
    const int mat = blockIdx.y >> 2; // 0=Q 1=K 2=V
    const int ct  = blockIdx.y & 3;  // 16-col tile of HS

    const float* W = (mat == 0) ? Wq : (mat == 1) ? Wk : Wv;
    float* out = qkv + (size_t)mat * QKV1;

    const float* arow = x + (size_t)(rt * 16 + nl) * EMB;  // this lane's A row
    const float* wcol = W + ct * 16 + nl;

    v8f a0 = {}, a1 = {}, a2 = {}, a3 = {};
    const int h2 = 2 * h;
    for (int s = 0; s < EMB / 4; s += 4) {
        a0 = proj_step(arow, wcol, 4 * (s + 0) + h2, a0);
        a1 = proj_step(arow, wcol, 4 * (s + 1) + h2, a1);
        a2 = proj_step(arow, wcol, 4 * (s + 2) + h2, a2);
        a3 = proj_step(arow, wcol, 4 * (s + 3) + h2, a3);
    }
    v8f acc = (a0 + a1) + (a2 + a3);

    // D layout: VGPR r -> row r + 8h, col ct*16 + nl
    float* orow = out + (size_t)(rt * 16 + 8 * h) * HS + ct * 16 + nl;
#pragma unroll
    for (int r = 0; r < 8; ++r) orow[(size_t)r * HS] = acc[r];
}

// grid: (B, T/16) ; block: 256 (8 waves). Each wave handles key tiles j = w, w+8, ...
__global__ __launch_bounds__(256)
void attn(const float* __restrict__ qkv, float* __restrict__ out) {
    const int tid  = threadIdx.x;
    const int lane = tid & 31;
    const int w    = tid >> 5;       // wave id 0..7
    const int nl   = lane & 15;
    const int h    = lane >> 4;
    const int h2   = 2 * h;
    const int b    = blockIdx.x;
    const int qt   = blockIdx.y;     // query tile
    const int q0   = qt * 16;

    const float* Qb = qkv + (size_t)b * TT * HS;
    const float* Kb = qkv + QKV1 + (size_t)b * TT * HS;
    const float* Vb = qkv + 2 * QKV1 + (size_t)b * TT * HS;

    __shared__ float p_lds[8][16 * 16];   // per-wave P staging (D->A relayout)
    __shared__ float o_lds[8][16 * HS];   // per-wave partial O
    __shared__ float m_lds[8][16];
    __shared__ float l_lds[8][16];

    // Q tile in A-layout fragments: lane holds row q0+nl, k = 4s + 2h (+1)
    v2f qa[16];
    const float* qrow = Qb + (size_t)(q0 + nl) * HS + h2;
#pragma unroll
    for (int s = 0; s < 16; ++s) qa[s] = *(const v2f*)(qrow + 4 * s);

    v8f O0 = {}, O1 = {}, O2 = {}, O3 = {};
    float mrow[8], lrow[8];
#pragma unroll
    for (int r = 0; r < 8; ++r) { mrow[r] = -1e30f; lrow[r] = 0.0f; }

    for (int j = w; j <= qt; j += 8) {
        // prefetch next K/V tile for this wave
        if (j + 8 <= qt) {
            __builtin_prefetch(Kb + (size_t)((j + 8) * 16 + nl) * HS + h2, 0, 1);
            __builtin_prefetch(Vb + (size_t)((j + 8) * 16 + nl) * HS + h2, 0, 1);
        }

        // K tile, B-layout (column n of K^T = row n of K): same pattern as Q
        const float* krow = Kb + (size_t)(j * 16 + nl) * HS + h2;
        v2f kb[16];
#pragma unroll
        for (int s = 0; s < 16; ++s) kb[s] = *(const v2f*)(krow + 4 * s);

        // S = Q @ K^T  (16 fp32 WMMAs, 2 accumulator chains)
        v8f sA = {}, sB = {};
#pragma unroll
        for (int s = 0; s < 16; s += 2) {
            sA = __builtin_amdgcn_wmma_f32_16x16x4_f32(false, qa[s],     false, kb[s],
                                                       (short)0, sA, false, false);
            sB = __builtin_amdgcn_wmma_f32_16x16x4_f32(false, qa[s + 1], false, kb[s + 1],
                                                       (short)0, sB, false, false);
        }
        v8f S = sA + sB;

        const bool diag = (j == qt);
        float scale_r[8];
#pragma unroll
        for (int r = 0; r < 8; ++r) {
            float sv = S[r];                       // row r+8h, key col j*16+nl
            if (diag && nl > r + 8 * h) sv = -1e30f;   // causal mask
            // row max within 16-lane half
            float mx = sv;
            mx = fmaxf(mx, __shfl_xor(mx, 1));
            mx = fmaxf(mx, __shfl_xor(mx, 2));
            mx = fmaxf(mx, __shfl_xor(mx, 4));
            mx = fmaxf(mx, __shfl_xor(mx, 8));
            const float mnew = fmaxf(mrow[r], mx);
            const float p = __expf(sv - mnew);
            float rs = p;
            rs += __shfl_xor(rs, 1);
            rs += __shfl_xor(rs, 2);
            rs += __shfl_xor(rs, 4);
            rs += __shfl_xor(rs, 8);
            const float sc = __expf(mrow[r] - mnew);
            lrow[r] = lrow[r] * sc + rs;
            mrow[r] = mnew;
            scale_r[r] = sc;
            p_lds[w][(r + 8 * h) * 16 + nl] = p;   // D-layout -> row-major stash
        }
#pragma unroll
        for (int r = 0; r < 8; ++r) {
            O0[r] *= scale_r[r]; O1[r] *= scale_r[r];
            O2[r] *= scale_r[r]; O3[r] *= scale_r[r];
        }

        // same-wave DS ops are in-order; just stop the compiler reordering
        asm volatile("" ::: "memory");
        v2f pa[4];
#pragma unroll
        for (int s = 0; s < 4; ++s)
            pa[s] = *(const v2f*)&p_lds[w][nl * 16 + 4 * s + h2];
        asm volatile("" ::: "memory");

        // O[t] += P @ V[:, 16t:16t+16]
        const float* vbase = Vb + (size_t)(j * 16) * HS;
#pragma unroll
        for (int s = 0; s < 4; ++s) {
            const int kk = 4 * s + h2;
            const float* vr = vbase + (size_t)kk * HS + nl;
            v2f b0, b1, b2, b3;
            b0.x = vr[0];  b0.y = vr[HS];
            b1.x = vr[16]; b1.y = vr[16 + HS];
            b2.x = vr[32]; b2.y = vr[32 + HS];
            b3.x = vr[48]; b3.y = vr[48 + HS];
            O0 = __builtin_amdgcn_wmma_f32_16x16x4_f32(false, pa[s], false, b0, (short)0, O0, false, false);
            O1 = __builtin_amdgcn_wmma_f32_16x16x4_f32(false, pa[s], false, b1, (short)0, O1, false, false);
            O2 = __builtin_amdgcn_wmma_f32_16x16x4_f32(false, pa[s], false, b2, (short)0, O2, false, false);
            O3 = __builtin_amdgcn_wmma_f32_16x16x4_f32(false, pa[s], false, b3, (short)0, O3, false, false);
        }
    }

    // dump per-wave partials
    if (nl == 0) {
#pragma unroll
        for (int r = 0; r < 8; ++r) {
            m_lds[w][r + 8 * h] = mrow[r];
            l_lds[w][r + 8 * h] = lrow[r];
        }
    }
    float* od = &o_lds[w][(8 * h) * HS + nl];
#pragma unroll
    for (int r = 0; r < 8; ++r) {
        od[r * HS +  0] = O0[r];
        od[r * HS + 16] = O1[r];
        od[r * HS + 32] = O2[r];
        od[r * HS + 48] = O3[r];
    }
    __syncthreads();

    // cross-wave log-sum-exp combine; final scale includes the reference's
    // softmax-then-/sqrt(HS) quirk (i.e. divide by 8).
    float* outb = out + ((size_t)b * TT + q0) * HS;
    for (int e = tid; e < 16 * HS; e += 256) {
        const int row = e >> 6, col = e & 63;
        float M = -1e30f;
#pragma unroll
        for (int wv = 0; wv < 8; ++wv) M = fmaxf(M, m_lds[wv][row]);
        float num = 0.0f, den = 0.0f;
#pragma unroll
        for (int wv = 0; wv < 8; ++wv) {
            const float f = __expf(m_lds[wv][row] - M);
            num += o_lds[wv][row * HS + col] * f;
            den += l_lds[wv][row] * f;
        }
        outb[(size_t)row * HS + col] = num / (den * 8.0f);
    }
}

extern "C" void kernel_launch(void* const* d_in, const int* in_sizes, int n_in,
                              void* d_out, int out_size, void* d_ws, size_t ws_size,
                              hipStream_t stream) {
    const float* x  = (const float*)d_in[0];
    const float* Wq = (const float*)d_in[1];
    const float* Wk = (const float*)d_in[2];
    const float* Wv = (const float*)d_in[3];
    float* outp = (float*)d_out;
    float* qkv  = (float*)d_ws;   // needs 3 * 16384 * 64 * 4 = 12 MB scratch

    dim3 g1(BT / 16, 12);
    qkv_proj<<<g1, 32, 0, stream>>>(x, Wq, Wk, Wv, qkv);

    dim3 g2(BB, TT / 16);
    attn<<<g2, 256, 0, stream>>>(qkv, outp);
}